// EdgeOperator_4784593568058
// MI455X (gfx1250) — compile-verified
//
#include <hip/hip_runtime.h>
#include <math.h>

#ifndef __has_builtin
#define __has_builtin(x) 0
#endif

#if __has_builtin(__builtin_amdgcn_global_load_async_to_lds_b32)
#define ASYNC_LDS 1
#else
#define ASYNC_LDS 0
#endif

#define AS1 __attribute__((address_space(1)))
#define AS3 __attribute__((address_space(3)))

// Gaussian 5-tap, sigma=1 (matches _gaussian_kernel1d(5,1.0))
constexpr double kE1 = 0.6065306597126334;   // exp(-0.5)
constexpr double kE2 = 0.1353352832366127;   // exp(-2)
constexpr double kS  = 1.0 + 2.0 * kE1 + 2.0 * kE2;
constexpr float W0 = (float)(kE2 / kS);
constexpr float W1 = (float)(kE1 / kS);
constexpr float W2 = (float)(1.0 / kS);

__device__ __forceinline__ int reflect512(int i) {
  i = (i < 0) ? -i : i;                // np.pad 'reflect' (no edge dup)
  return (i > 511) ? (1022 - i) : i;
}
__device__ __forceinline__ int clamp511(int i) {
  return (i < 0) ? 0 : ((i > 511) ? 511 : i);
}

__global__ __launch_bounds__(256) void canny_fused(const float* __restrict__ in,
                                                   float* __restrict__ out) {
  const int ox = blockIdx.x * 32;
  const int oy = blockIdx.y * 32;
  const int b  = blockIdx.z;
  const int tid = threadIdx.x;

  __shared__ float s_chan[3][1600];  // 40x40 raw channels (reflect-indexed)
  __shared__ float s_gray[1600];     // 40x40 grayscale
  __shared__ float s_hb[1440];       // 40 rows x 36 cols, horizontal blur
  __shared__ float s_bl[1296];       // 36x36 blurred (edge-clamp baked in)
  __shared__ float s_mag[1156];      // 34x34 magnitude (zero outside image)

  // ---- Phase A: stage 3 channels of the 40x40 halo tile via async-to-LDS ----
  const size_t base = (size_t)b * 3u * 512u * 512u;
  for (int s = tid; s < 1600; s += 256) {
    const int ly = s / 40, lx = s - ly * 40;
    const int ry = reflect512(oy + ly - 4);
    const int rx = reflect512(ox + lx - 4);
    const size_t off = base + (size_t)ry * 512 + rx;
#if ASYNC_LDS
    __builtin_amdgcn_global_load_async_to_lds_b32(
        (AS1 int*)(in + off),             (AS3 int*)&s_chan[0][s], 0, 0);
    __builtin_amdgcn_global_load_async_to_lds_b32(
        (AS1 int*)(in + off + 262144u),   (AS3 int*)&s_chan[1][s], 0, 0);
    __builtin_amdgcn_global_load_async_to_lds_b32(
        (AS1 int*)(in + off + 524288u),   (AS3 int*)&s_chan[2][s], 0, 0);
#else
    s_chan[0][s] = in[off];
    s_chan[1][s] = in[off + 262144u];
    s_chan[2][s] = in[off + 524288u];
#endif
  }
#if ASYNC_LDS
#if __has_builtin(__builtin_amdgcn_s_wait_asynccnt)
  __builtin_amdgcn_s_wait_asynccnt(0);
#else
  asm volatile("s_wait_asynccnt 0" ::: "memory");
#endif
#endif
  __syncthreads();

  // ---- Phase B: grayscale of (data+1)/2 ----
  for (int s = tid; s < 1600; s += 256) {
    const float xr = (s_chan[0][s] + 1.0f) * 0.5f;
    const float xg = (s_chan[1][s] + 1.0f) * 0.5f;
    const float xb = (s_chan[2][s] + 1.0f) * 0.5f;
    s_gray[s] = 0.299f * xr + 0.587f * xg + 0.114f * xb;
  }
  __syncthreads();

  // ---- Phase C: horizontal 5-tap blur (x edge-clamp baked in) ----
  for (int s = tid; s < 1440; s += 256) {
    const int r = s / 36, u = s - r * 36;
    const int xc = clamp511(ox + u - 2);
    const int c0 = xc - ox + 2;  // gray slot of image col (xc-2)
    const float* g = &s_gray[r * 40 + c0];
    s_hb[s] = W0 * (g[0] + g[4]) + W1 * (g[1] + g[3]) + W2 * g[2];
  }
  __syncthreads();

  // ---- Phase D: vertical 5-tap blur (y edge-clamp baked in) ----
  for (int s = tid; s < 1296; s += 256) {
    const int ty = s / 36, tx = s - ty * 36;
    const int yc = clamp511(oy + ty - 2);
    const int r0 = yc - oy + 2;  // hblur row slot of image row (yc-2)
    const float* h = &s_hb[r0 * 36 + tx];
    s_bl[s] = W0 * (h[0] + h[4 * 36]) + W1 * (h[36] + h[3 * 36]) + W2 * h[2 * 36];
  }
  __syncthreads();

  // ---- Phase E: Sobel + magnitude on 34x34 (zero outside image for NMS pad) ----
  for (int s = tid; s < 1156; s += 256) {
    const int my = s / 34, mx = s - my * 34;
    const int iy = oy + my - 1, ix = ox + mx - 1;
    float m = 0.0f;
    if ((unsigned)iy < 512u && (unsigned)ix < 512u) {
      const float* p = &s_bl[my * 36 + mx];  // top-left neighbor of center
      const float a00 = p[0],  a01 = p[1],  a02 = p[2];
      const float a10 = p[36],               a12 = p[38];
      const float a20 = p[72], a21 = p[73], a22 = p[74];
      const float gx = (a02 - a00) + 2.0f * (a12 - a10) + (a22 - a20);
      const float gy = (a20 - a00) + 2.0f * (a21 - a01) + (a22 - a02);
      m = sqrtf(gx * gx + gy * gy + 1e-6f);
    }
    s_mag[s] = m;
  }
  __syncthreads();

  // ---- Phase F: directional NMS + store ----
  const int tx = tid & 31;
  const int ty0 = tid >> 5;
  for (int rr = 0; rr < 4; ++rr) {
    const int py = ty0 + rr * 8;
    const int iy = oy + py, ix = ox + tx;
    const float mc = s_mag[(py + 1) * 34 + (tx + 1)];
    const float* p = &s_bl[(py + 1) * 36 + (tx + 1)];
    const float a00 = p[0],  a01 = p[1],  a02 = p[2];
    const float a10 = p[36],               a12 = p[38];
    const float a20 = p[72], a21 = p[73], a22 = p[74];
    const float gx = (a02 - a00) + 2.0f * (a12 - a10) + (a22 - a20);
    const float gy = (a20 - a00) + 2.0f * (a21 - a01) + (a22 - a02);
    // ang45 = round(atan2*4/pi), idx = mod 8 (Python semantics)
    const float t = rintf(atan2f(gy, gx) * 1.27323954473516268f);
    const int idx = ((int)t + 8) & 7;
    // packed 2-bit (d+1) tables: dy: {0,-1,-1,-1,0,1,1,1}, dx: {1,1,0,-1,-1,-1,0,1}
    const int dy = ((0xA901 >> (2 * idx)) & 3) - 1;
    const int dx = ((0x901A >> (2 * idx)) & 3) - 1;
    const float sp = mc - s_mag[(py + 1 + dy) * 34 + (tx + 1 + dx)];
    const float sn = mc - s_mag[(py + 1 - dy) * 34 + (tx + 1 - dx)];
    out[((size_t)b * 512 + iy) * 512 + ix] = (fminf(sp, sn) > 0.0f) ? mc : 0.0f;
  }
}

extern "C" void kernel_launch(void* const* d_in, const int* in_sizes, int n_in,
                              void* d_out, int out_size, void* d_ws, size_t ws_size,
                              hipStream_t stream) {
  (void)in_sizes; (void)n_in; (void)out_size; (void)d_ws; (void)ws_size;
  const float* in = (const float*)d_in[0];
  float* out = (float*)d_out;
  dim3 grid(16, 16, 16);  // 512/32 x 512/32 tiles, 16 batches
  canny_fused<<<grid, dim3(256), 0, stream>>>(in, out);
}